// GCNOpt_21114059227154
// MI455X (gfx1250) — compile-verified
//
#include <hip/hip_runtime.h>

// ---------------------------------------------------------------------------
// 3-layer GCN for MI455X (gfx1250, wave32).
//   h = relu( in_norm * A * (out_norm * x) @ W )  x3 (last layer no relu)
// GEMMs: V_WMMA_F32_16X16X4_F32, one wave owns a full 16-row x N strip
// (NT accumulators) so the A fragment + row-scale is loaded once per K-step
// and reused by NT WMMAs. W is pre-packed so B fragments are single b64 loads.
// SpMM: one 32-lane wave per edge row, float4 per lane, global_atomic_add_f32
// (100Kx128 f32 = 51 MB, resident in the 192 MB L2).
// ---------------------------------------------------------------------------

typedef float v2f __attribute__((ext_vector_type(2)));
typedef float v8f __attribute__((ext_vector_type(8)));

__global__ __launch_bounds__(256) void zero4_kernel(uint4* __restrict__ p, long n4) {
  long i = (long)blockIdx.x * blockDim.x + threadIdx.x;
  long stride = (long)gridDim.x * blockDim.x;
  uint4 z; z.x = 0u; z.y = 0u; z.z = 0u; z.w = 0u;
  for (; i < n4; i += stride) p[i] = z;
}

__global__ __launch_bounds__(256) void degree_kernel(const int* __restrict__ src,
                                                     const int* __restrict__ dst,
                                                     int* __restrict__ degOut,
                                                     int* __restrict__ degIn, int nE) {
  int e = blockIdx.x * blockDim.x + threadIdx.x;
  if (e < nE) {
    atomicAdd(&degOut[src[e]], 1);
    atomicAdd(&degIn[dst[e]], 1);
  }
}

__global__ __launch_bounds__(256) void norm_kernel(const int* __restrict__ degOut,
                                                   const int* __restrict__ degIn,
                                                   float* __restrict__ outNorm,
                                                   float* __restrict__ inNorm,
                                                   float* __restrict__ comb, int n) {
  int i = blockIdx.x * blockDim.x + threadIdx.x;
  if (i < n) {
    float on = rsqrtf(fmaxf((float)degOut[i], 1.0f));
    float in = rsqrtf(fmaxf((float)degIn[i], 1.0f));
    outNorm[i] = on;
    inNorm[i]  = in;
    comb[i]    = on * in;   // relu(m*in)*out == relu(m)*(in*out), norms > 0
  }
}

// Pack W[K,N] row-major into K-pair-interleaved Wp so a lane's B fragment
// {W[k][n], W[k+1][n]} is one aligned 8-byte load:
//   Wp[(k/2)*N + n] = float2{ W[k][n], W[k+1][n] }
__global__ __launch_bounds__(256) void pack_w_kernel(const float* __restrict__ W,
                                                     float* __restrict__ Wp,
                                                     int K, int N) {
  int i = blockIdx.x * blockDim.x + threadIdx.x;   // over (K/2)*N
  int total = (K >> 1) * N;
  if (i < total) {
    int kp = i / N;
    int n  = i - kp * N;
    Wp[2 * i + 0] = W[(size_t)(2 * kp + 0) * N + n];
    Wp[2 * i + 1] = W[(size_t)(2 * kp + 1) * N + n];
  }
}

// Y[M, NT*16] = (scaleA[m] * maybe_relu(X[m,k])) @ W[k,n]
// One wave computes a 16 x (NT*16) strip: NT v8f accumulators.
// V_WMMA_F32_16X16X4_F32 fragment layout (ISA 7.12.2):
//   A (16x4):  lane = m + 16*(k>=2), vgpr = k&1
//   B (4x16):  lane = n + 16*(k>=2), vgpr = k&1   (mirror of A)
//   C/D (16x16): vgpr r -> row r + 8*(lane>=16), col = lane&15
template<int RELU, int NT>
__global__ __launch_bounds__(256) void gemm_wmma_f32(const float* __restrict__ X,
                                                     const float* __restrict__ Wp,
                                                     const float* __restrict__ scaleA,
                                                     float* __restrict__ Y,
                                                     int M, int K) {
  const int N     = NT * 16;
  const int lane  = threadIdx.x & 31;
  const int wave  = threadIdx.x >> 5;
  const int tileM = blockIdx.x * 8 + wave;
  if (tileM * 16 >= M) return;           // wave-uniform: EXEC stays all-ones

  const int m16   = lane & 15;           // row (A) / col-in-tile (B)
  const int khalf = lane >> 4;           // 0 -> K pair {0,1}, 1 -> {2,3}
  const int rowA  = tileM * 16 + m16;

  const float s = scaleA[rowA];
  const float* __restrict__ xrow  = X  + (size_t)rowA * K + 2 * khalf;
  // packed B base for this lane: pair index khalf within K-quad, column m16
  const float* __restrict__ wpair = Wp + ((size_t)khalf * N + m16) * 2;

  v8f zero = {};
  v8f acc[NT];
  #pragma unroll
  for (int t = 0; t < NT; ++t) acc[t] = zero;

  #pragma unroll 2
  for (int kb = 0; kb < K; kb += 4) {
    v2f a = *(const v2f*)(xrow + kb);          // 8B aligned
    if (RELU) { a.x = fmaxf(a.x, 0.0f); a.y = fmaxf(a.y, 0.0f); }
    a.x *= s; a.y *= s;
    const float* __restrict__ wk = wpair + (size_t)(kb >> 1) * N * 2;
    #pragma unroll
    for (int t = 0; t < NT; ++t) {
      v2f b = *(const v2f*)(wk + t * 32);      // {W[k][n], W[k+1][n]}
      acc[t] = __builtin_amdgcn_wmma_f32_16x16x4_f32(
          /*neg_a=*/false, a, /*neg_b=*/false, b,
          /*c_mod=*/(short)0, acc[t], /*reuse_a=*/false, /*reuse_b=*/false);
    }
  }

  #pragma unroll
  for (int t = 0; t < NT; ++t) {
    float* __restrict__ yout =
        Y + ((size_t)tileM * 16 + 8 * khalf) * N + (size_t)t * 16 + m16;
    #pragma unroll
    for (int r = 0; r < 8; ++r)
      yout[(size_t)r * N] = acc[t][r];
  }
}

// agg[dst] += Y[src]; tpe = C/4 lanes per edge, one float4 per lane.
__global__ __launch_bounds__(256) void spmm_kernel(const float* __restrict__ Y,
                                                   const int* __restrict__ srcI,
                                                   const int* __restrict__ dstI,
                                                   float* __restrict__ agg,
                                                   int nE, int C, int tpeShift) {
  const int tpe = 1 << tpeShift;
  const int epb = 256 >> tpeShift;
  const int e = blockIdx.x * epb + (threadIdx.x >> tpeShift);
  if (e >= nE) return;
  const int c = (threadIdx.x & (tpe - 1)) << 2;
  const int s = srcI[e];
  const int d = dstI[e];
  const float4 v = *(const float4*)(Y + (size_t)s * C + c);
  float* o = agg + (size_t)d * C + c;
  atomicAdd(o + 0, v.x);
  atomicAdd(o + 1, v.y);
  atomicAdd(o + 2, v.z);
  atomicAdd(o + 3, v.w);
}

// out = agg * in_norm (final layer: no relu), C = 2^cShift columns.
__global__ __launch_bounds__(256) void final_scale_kernel(const float* __restrict__ agg,
                                                          const float* __restrict__ inNorm,
                                                          float* __restrict__ out,
                                                          long total, int cShift) {
  long i = (long)blockIdx.x * blockDim.x + threadIdx.x;
  if (i < total) out[i] = agg[i] * inNorm[i >> cShift];
}

static inline int ceil_div(long a, long b) { return (int)((a + b - 1) / b); }

extern "C" void kernel_launch(void* const* d_in, const int* in_sizes, int n_in,
                              void* d_out, int out_size, void* d_ws, size_t ws_size,
                              hipStream_t stream) {
  const float* features = (const float*)d_in[0];   // [N,256]
  const int*   src      = (const int*)d_in[1];     // [E]
  const int*   dst      = (const int*)d_in[2];     // [E]
  const float* W0       = (const float*)d_in[3];   // [256,128]
  const float* W1       = (const float*)d_in[4];   // [128,128]
  const float* W2       = (const float*)d_in[5];   // [128,64]

  const int nNodes = in_sizes[0] / 256;
  const int nEdges = in_sizes[1];

  // Workspace carve-out (256B aligned).
  char* ws = (char*)d_ws;
  size_t off = 0;
  auto carve = [&](size_t bytes) {
    void* p = ws + off;
    off = (off + bytes + 255) & ~(size_t)255;
    return p;
  };
  int*   degOut  = (int*)  carve((size_t)nNodes * 4);
  int*   degIn   = (int*)  carve((size_t)nNodes * 4);
  float* outNorm = (float*)carve((size_t)nNodes * 4);
  float* inNorm  = (float*)carve((size_t)nNodes * 4);
  float* comb    = (float*)carve((size_t)nNodes * 4);
  float* Wp0     = (float*)carve((size_t)256 * 128 * 4);     // packed weights
  float* Wp1     = (float*)carve((size_t)128 * 128 * 4);
  float* Wp2     = (float*)carve((size_t)128 * 64 * 4);
  float* ybuf    = (float*)carve((size_t)nNodes * 128 * 4);  // GEMM output
  float* agg     = (float*)carve((size_t)nNodes * 128 * 4);  // SpMM accumulator

  const int ZB = 4096;  // blocks for zero-fill (grid-stride)

  // --- degrees, norms, weight packing -------------------------------------
  zero4_kernel<<<ZB, 256, 0, stream>>>((uint4*)degOut, (long)nNodes / 4 + 1);
  zero4_kernel<<<ZB, 256, 0, stream>>>((uint4*)degIn,  (long)nNodes / 4 + 1);
  degree_kernel<<<ceil_div(nEdges, 256), 256, 0, stream>>>(src, dst, degOut, degIn, nEdges);
  norm_kernel<<<ceil_div(nNodes, 256), 256, 0, stream>>>(degOut, degIn, outNorm, inNorm, comb, nNodes);
  pack_w_kernel<<<ceil_div(128 * 128, 256), 256, 0, stream>>>(W0, Wp0, 256, 128);
  pack_w_kernel<<<ceil_div(64 * 128, 256), 256, 0, stream>>>(W1, Wp1, 128, 128);
  pack_w_kernel<<<ceil_div(64 * 64, 256), 256, 0, stream>>>(W2, Wp2, 128, 64);

  const int tilesM   = ceil_div(nNodes, 16);
  const int gemmBlks = ceil_div(tilesM, 8);   // 8 waves/block, 1 row-strip/wave

  // --- layer 0: y = (out_norm*x) @ W0 ; agg = A*y --------------------------
  gemm_wmma_f32<0, 8><<<gemmBlks, 256, 0, stream>>>(features, Wp0, outNorm, ybuf, nNodes, 256);
  zero4_kernel<<<ZB, 256, 0, stream>>>((uint4*)agg, (long)nNodes * 128 / 4);
  spmm_kernel<<<ceil_div(nEdges, 8), 256, 0, stream>>>(ybuf, src, dst, agg, nEdges, 128, 5);

  // --- layer 1: y = (comb*relu(agg)) @ W1 ; agg = A*y ----------------------
  gemm_wmma_f32<1, 8><<<gemmBlks, 256, 0, stream>>>(agg, Wp1, comb, ybuf, nNodes, 128);
  zero4_kernel<<<ZB, 256, 0, stream>>>((uint4*)agg, (long)nNodes * 128 / 4);
  spmm_kernel<<<ceil_div(nEdges, 8), 256, 0, stream>>>(ybuf, src, dst, agg, nEdges, 128, 5);

  // --- layer 2: y = (comb*relu(agg)) @ W2 ; out = in_norm * (A*y) ----------
  gemm_wmma_f32<1, 4><<<gemmBlks, 256, 0, stream>>>(agg, Wp2, comb, ybuf, nNodes, 128);
  zero4_kernel<<<ZB, 256, 0, stream>>>((uint4*)agg, (long)nNodes * 64 / 4);
  spmm_kernel<<<ceil_div(nEdges, 16), 256, 0, stream>>>(ybuf, src, dst, agg, nEdges, 64, 4);
  final_scale_kernel<<<ceil_div((long)nNodes * 64, 256), 256, 0, stream>>>(
      agg, inNorm, (float*)d_out, (long)nNodes * 64, 6);
}